// L2CEmbedding_79559974191444
// MI455X (gfx1250) — compile-verified
//
#include <hip/hip_runtime.h>
#include <stdint.h>

#define BATCH   4096
#define IN_DIM  23
#define CTX_L   20
#define NSENSE  5
#define DIM     128
#define EPS     0.1f
#define WPB     8                 // waves per block (1 row per wave)
#define NBUF    5                 // LDS pipeline depth per wave
#define SLICE   (NSENSE * DIM * 4)   // 2560 B: one word's W_s[w] slice
#define ROWB    (DIM * 4)            // 512 B: one W_g row

typedef __attribute__((ext_vector_type(2))) float v2f;
typedef __attribute__((ext_vector_type(4))) float v4f;
typedef __attribute__((ext_vector_type(8))) float v8f;
typedef __attribute__((ext_vector_type(4))) int   v4i;

// ---- async global -> LDS (CDNA5), tracked with ASYNCcnt ----------------
__device__ __forceinline__ void async_copy_b128(const void* g, void* l) {
#if __has_builtin(__builtin_amdgcn_global_load_async_to_lds_b128)
  __builtin_amdgcn_global_load_async_to_lds_b128(
      (__attribute__((address_space(1))) v4i*)g,
      (__attribute__((address_space(3))) v4i*)l, /*offset=*/0, /*cpol=*/0);
#else
  unsigned loff = (unsigned)(uintptr_t)(__attribute__((address_space(3))) char*)(char*)l;
  unsigned long long ga = (unsigned long long)(uintptr_t)g;
  asm volatile("global_load_async_to_lds_b128 %0, %1, off"
               :: "v"(loff), "v"(ga) : "memory");
#endif
}

template <int N>
__device__ __forceinline__ void wait_async() {
#if __has_builtin(__builtin_amdgcn_s_wait_asynccnt)
  __builtin_amdgcn_s_wait_asynccnt(N);
#else
  asm volatile("s_wait_asynccnt %0" :: "i"(N) : "memory");
#endif
  asm volatile("" ::: "memory");   // keep LDS reads below the wait
}

__device__ __forceinline__ void wait_async_n(int pend) {
  // pend is compile-time constant after full unroll: 0,5,10,15,20
  switch (pend) {
    case 0:  wait_async<0>();  break;
    case 5:  wait_async<5>();  break;
    case 10: wait_async<10>(); break;
    case 15: wait_async<15>(); break;
    default: wait_async<20>(); break;
  }
}

// ---- full-wave sum of (pa + pb) using the matrix pipe ------------------
// A(16x4 f32) = [pa, pb] per lane; B = ones  =>  D[m,*] = pa(m)+pb(m)+pa(m+16)+pb(m+16).
// Each lane then holds 8 of the 16 row-sums in the accumulator; 7 adds +
// one shfl_xor(16) give the full 32-lane sum broadcast to every lane.
__device__ __forceinline__ float wave_sum2(float pa, float pb) {
#if __has_builtin(__builtin_amdgcn_wmma_f32_16x16x4_f32)
  v2f a; a[0] = pa;  a[1] = pb;
  v2f b; b[0] = 1.f; b[1] = 1.f;
  v8f c = {};
  c = __builtin_amdgcn_wmma_f32_16x16x4_f32(false, a, false, b, (short)0, c,
                                            false, false);
  float s = ((c[0] + c[1]) + (c[2] + c[3])) + ((c[4] + c[5]) + (c[6] + c[7]));
  s += __shfl_xor(s, 16, 32);
  return s;
#else
  float s = pa + pb;
  #pragma unroll
  for (int off = 16; off > 0; off >>= 1) s += __shfl_xor(s, off, 32);
  return s;
#endif
}

__global__ __launch_bounds__(WPB * 32, 1)
void l2c_embedding_kernel(const int* __restrict__ x,
                          const float* __restrict__ Wg,
                          const float* __restrict__ Ws,
                          float* __restrict__ out) {
  __shared__ __align__(16) char smem[WPB * NBUF * SLICE];   // 100 KB / block

  const int lane = threadIdx.x & 31;
  const int wave = threadIdx.x >> 5;
  const int row  = blockIdx.x * WPB + wave;
  if (row >= BATCH) return;                 // wave-uniform (never splits EXEC)

  char*       buf0 = smem + wave * (NBUF * SLICE);
  const int*  xr   = x + row * IN_DIM;
  const char* WgB  = (const char*)Wg;
  const char* WsB  = (const char*)Ws;
  const int   lb   = lane * 16;             // this lane's float4 byte offset

  int ctx[CTX_L];
  #pragma unroll
  for (int j = 0; j < CTX_L; ++j) ctx[j] = xr[3 + j];

  // ---- prologue: async-stage W_s slices for steps 0..NBUF-2 ----
  #pragma unroll
  for (int p = 0; p < NBUF - 1; ++p) {
    const char* g = WsB + (size_t)ctx[p] * SLICE + lb;
    char*       l = buf0 + p * SLICE + lb;
    #pragma unroll
    for (int s = 0; s < NSENSE; ++s)
      async_copy_b128(g + s * 512, l + s * 512);
  }

  // ---- cv0 = sum of W_g[context] (overlaps the async staging) ----
  v4f cv = {0.f, 0.f, 0.f, 0.f};
  #pragma unroll
  for (int j = 0; j < CTX_L; ++j)
    cv += *(const v4f*)(WgB + (size_t)ctx[j] * ROWB + lb);

  unsigned long long pack = 0ull;           // 3 bits per step: chosen sense

  // ---- serial disambiguation chain, software-pipelined ----
  #pragma unroll
  for (int i = 0; i < CTX_L; ++i) {
    if (i + NBUF - 1 < CTX_L) {             // stage step i+4
      const char* g = WsB + (size_t)ctx[i + NBUF - 1] * SLICE + lb;
      char*       l = buf0 + ((i + NBUF - 1) % NBUF) * SLICE + lb;
      #pragma unroll
      for (int s = 0; s < NSENSE; ++s)
        async_copy_b128(g + s * 512, l + s * 512);
    }
    const int last = (i + NBUF - 1 < CTX_L - 1) ? (i + NBUF - 1) : (CTX_L - 1);
    wait_async_n(5 * (last - i));           // step i's 5 loads have landed

    const char* bi = buf0 + (i % NBUF) * SLICE;
    float sc[NSENSE];
    #pragma unroll
    for (int s = 0; s < NSENSE; ++s) {      // scores = W_s[w] @ cv
      v4f w4 = *(const v4f*)(bi + s * 512 + lb);
      float pa = fmaf(w4[0], cv[0], w4[1] * cv[1]);
      float pb = fmaf(w4[2], cv[2], w4[3] * cv[3]);
      sc[s] = wave_sum2(pa, pb);            // v_wmma reduction
    }

    // top-2 (stable: strict '>' keeps lowest index on ties, like lax.top_k)
    float v0 = sc[0]; int bidx = 0;
    #pragma unroll
    for (int s = 1; s < NSENSE; ++s) if (sc[s] > v0) { v0 = sc[s]; bidx = s; }
    float v1 = -3.4e38f;
    #pragma unroll
    for (int s = 0; s < NSENSE; ++s) if (s != bidx && sc[s] > v1) v1 = sc[s];

    // prev sense is always -1 here, so the swap is  cv - W_g[w] + W_s[w,best]
    v4f wsb = *(const v4f*)(bi + bidx * 512 + lb);
    v4f wg  = *(const v4f*)(WgB + (size_t)ctx[i] * ROWB + lb);
    v4f ncv = cv + wsb - wg;
    cv = (v0 - v1 > EPS) ? ncv : cv;

    pack |= (unsigned long long)bidx << (3 * i);
  }

  // ---- epilogue: sigmoid(sense_dot + global_dot) ----
  const int x0 = xr[0], x1 = xr[1], inx = xr[2];
  const int s13 = (int)((pack >> (3 * 13)) & 7);          // actual_word_indx
  const int cl  = inx < 0 ? 0 : inx;
  const int srh = (int)((pack >> (3 * cl)) & 7);
  const int sidx = (inx <= -1) ? (-inx - 1) : srh;

  v4f wsv = *(const v4f*)(WsB + ((size_t)x0 * NSENSE + s13) * 512 + lb);
  const char* cb = (sidx < 0) ? (WgB + (size_t)x0 * ROWB)
                              : (WsB + ((size_t)x0 * NSENSE + sidx) * 512);
  v4f csv = *(const v4f*)(cb + lb);
  v4f g0  = *(const v4f*)(WgB + (size_t)x0 * ROWB + lb);
  v4f g1  = *(const v4f*)(WgB + (size_t)x1 * ROWB + lb);

  float pa = fmaf(wsv[0], csv[0], wsv[1] * csv[1]) +
             fmaf(wsv[2], csv[2], wsv[3] * csv[3]);
  float pb = fmaf(g0[0], g1[0], g0[1] * g1[1]) +
             fmaf(g0[2], g1[2], g0[3] * g1[3]);
  float t = wave_sum2(pa, pb);
  if (lane == 0) out[row] = 1.0f / (1.0f + __expf(-t));
}

extern "C" void kernel_launch(void* const* d_in, const int* in_sizes, int n_in,
                              void* d_out, int out_size, void* d_ws, size_t ws_size,
                              hipStream_t stream) {
  const int*   x  = (const int*)d_in[0];
  const float* Wg = (const float*)d_in[1];
  const float* Ws = (const float*)d_in[2];
  float* out = (float*)d_out;
  (void)in_sizes; (void)n_in; (void)out_size; (void)d_ws; (void)ws_size;

  dim3 grid((BATCH + WPB - 1) / WPB);   // 512 blocks
  dim3 block(WPB * 32);                 // 8 wave32 waves
  hipLaunchKernelGGL(l2c_embedding_kernel, grid, block, 0, stream, x, Wg, Ws, out);
}